// GConv_1288490189508
// MI455X (gfx1250) — compile-verified
//
#include <hip/hip_runtime.h>
#include <hip/hip_bf16.h>

// ---------------------------------------------------------------------------
// GCN forward on MI455X (gfx1250), fp32 throughout to match the reference.
//   g  = (X @ W) * dinv[row]          (WMMA f32 16x16x4 GEMM, fused scale)
//   acc[c] = g[c] + sum_{e: col=c} g[row_e]    (L2-resident float atomics)
//   z  = prelu(dinv[c]*acc[c] + b)
// Two layers, then batchnorm -> z (out0), then z@projW+b -> BN -> prelu -> p.
//
// Weights are pre-packed once into K-pair-interleaved layout
//   PW[kb/2][n][j] = W[kb+j][n]   (kb even, j in {0,1})
// so every WMMA B fragment is one aligned global_load_b64 whose 16-lane
// half-wave footprint is a contiguous 128B segment (single L2 transaction).
// ---------------------------------------------------------------------------

typedef __attribute__((ext_vector_type(2))) float v2f;
typedef __attribute__((ext_vector_type(8))) float v8f;

__global__ __launch_bounds__(256) void fill_f32(float* __restrict__ p, float v, int n) {
  int i = blockIdx.x * 256 + threadIdx.x;
  if (i < n) p[i] = v;
}

__global__ __launch_bounds__(256) void deg_kernel(const int* __restrict__ ei,
                                                  float* __restrict__ deg, int E) {
  int e = blockIdx.x * 256 + threadIdx.x;
  if (e < E) atomicAdd(&deg[ei[E + e]], 1.0f);
}

__global__ __launch_bounds__(256) void rsqrt_kernel(const float* __restrict__ deg,
                                                    float* __restrict__ dinv, int n) {
  int i = blockIdx.x * 256 + threadIdx.x;
  if (i < n) dinv[i] = rsqrtf(deg[i]);
}

// Pack 256x256 row-major W into K-pair-interleaved PW (65536 floats).
// PW[p*512 + 2n + j] = W[(2p + j)*256 + n]
__global__ __launch_bounds__(256) void pack_w(const float* __restrict__ W,
                                              float* __restrict__ PW) {
  const int idx = blockIdx.x * 256 + threadIdx.x;  // 128 p-pairs x 256 n
  if (idx >= 128 * 256) return;
  const int p = idx >> 8;
  const int n = idx & 255;
  PW[(size_t)p * 512 + 2 * n + 0] = W[(size_t)(2 * p + 0) * 256 + n];
  PW[(size_t)p * 512 + 2 * n + 1] = W[(size_t)(2 * p + 1) * 256 + n];
}

// ---------------------------------------------------------------------------
// GEMM: out[M x 256] = A[M x 256] @ W[256 x 256], fp32 WMMA 16x16x4.
// MODE 0: v *= vec[row]  (vec = dinv), dual-write out0 (G) and out1 (ACC seed)
// MODE 1: v += vec[col]  (vec = bias), write out0 only
// Block = 256 threads = 8 waves; block tile = 32 rows x 256 cols.
// Wave (w): row_tile = w>>2 (16 rows), col stripe = (w&3)*64 (4 accum tiles).
// LDS A tile padded to stride 260 -> A-fragment ds reads hit all 64 banks.
// ---------------------------------------------------------------------------
template <int MODE>
__global__ __launch_bounds__(256) void gemm256(const float* __restrict__ A,
                                               const float* __restrict__ PW,
                                               const float* __restrict__ vec,
                                               float* __restrict__ out0,
                                               float* __restrict__ out1, int M) {
  __shared__ float As[32][260];
  const int tid = threadIdx.x;
  const int row_base = blockIdx.x * 32;

  // Cooperative float4 load of the 32x256 A tile (zero-pad past M).
  for (int i = tid; i < 32 * 64; i += 256) {
    const int r = i >> 6;
    const int c4 = (i & 63) << 2;
    const int grow = row_base + r;
    float4 v = make_float4(0.f, 0.f, 0.f, 0.f);
    if (grow < M) v = *reinterpret_cast<const float4*>(A + (size_t)grow * 256 + c4);
    As[r][c4 + 0] = v.x; As[r][c4 + 1] = v.y;
    As[r][c4 + 2] = v.z; As[r][c4 + 3] = v.w;
  }
  __syncthreads();

  const int wv = tid >> 5;
  const int lane = tid & 31;
  const int lmod = lane & 15;        // N index within tile / M index for A frag
  const int lhalf = lane >> 4;       // selects K pair (A/B) or M+8 (C/D)
  const int row_tile = wv >> 2;      // 0..1
  const int ncol0 = (wv & 3) * 64;   // 0,64,128,192
  const int arow = row_tile * 16 + lmod;

  v8f acc[4];
#pragma unroll
  for (int i = 0; i < 4; ++i)
#pragma unroll
    for (int j = 0; j < 8; ++j) acc[i][j] = 0.0f;

  // Per-lane base into packed W: pair index (k>>1)+lhalf, column (ncol0+lmod).
  const float* Wbase = PW + ((size_t)lhalf << 9) + ((size_t)(ncol0 + lmod) << 1);

  for (int k = 0; k < 256; k += 4) {
    const int kb = k + 2 * lhalf;
    v2f af;
    af[0] = As[arow][kb];
    af[1] = As[arow][kb + 1];
    const float* Wp = Wbase + ((size_t)(k >> 1) << 9);
#pragma unroll
    for (int nt = 0; nt < 4; ++nt) {
      const v2f bf = *reinterpret_cast<const v2f*>(Wp + (nt << 5));  // b64 load
      acc[nt] = __builtin_amdgcn_wmma_f32_16x16x4_f32(
          false, af, false, bf, (short)0, acc[nt], false, false);
    }
  }

  // Epilogue: C/D layout -> row = base + r + 8*lhalf, col = ncol0 + nt*16 + lmod
#pragma unroll
  for (int nt = 0; nt < 4; ++nt) {
    const int col = ncol0 + nt * 16 + lmod;
#pragma unroll
    for (int r = 0; r < 8; ++r) {
      const int row = row_base + row_tile * 16 + lhalf * 8 + r;
      if (row < M) {
        float v = acc[nt][r];
        const size_t o = (size_t)row * 256 + col;
        if (MODE == 0) {
          v *= vec[row];
          out0[o] = v;  // G (gather source)
          out1[o] = v;  // ACC seeded with self-loop term
        } else {
          v += vec[col];
          out0[o] = v;
        }
      }
    }
  }
}

// Edge scatter: ACC[col] += G[row]; 64 lanes x float4 per edge, 4 edges/block.
__global__ __launch_bounds__(256) void scatter_kernel(const float* __restrict__ G,
                                                      float* __restrict__ ACC,
                                                      const int* __restrict__ ei, int E) {
  const int e = blockIdx.x * 4 + (threadIdx.x >> 6);
  const int le = threadIdx.x & 63;
  if (e >= E) return;
  const int row = ei[e];
  const int col = ei[E + e];
  const float4 g = *reinterpret_cast<const float4*>(G + (size_t)row * 256 + le * 4);
  float* a = ACC + (size_t)col * 256 + le * 4;
  atomicAdd(a + 0, g.x);
  atomicAdd(a + 1, g.y);
  atomicAdd(a + 2, g.z);
  atomicAdd(a + 3, g.w);
}

// z[i,f] = prelu(dinv[i]*ACC[i,f] + b[f])
__global__ __launch_bounds__(256) void gcn_epilogue(const float* __restrict__ ACC,
                                                    const float* __restrict__ dinv,
                                                    const float* __restrict__ bias,
                                                    const float* __restrict__ alpha_p,
                                                    float* __restrict__ Z, int n) {
  const int i = blockIdx.x * 256 + threadIdx.x;  // over n*64 float4 groups
  if (i >= n * 64) return;
  const int row = i >> 6;
  const int c4 = (i & 63) << 2;
  const float a = *alpha_p;
  const float di = dinv[row];
  float4 v = *reinterpret_cast<const float4*>(ACC + (size_t)row * 256 + c4);
  const float4 b = *reinterpret_cast<const float4*>(bias + c4);
  v.x = v.x * di + b.x; v.x = (v.x >= 0.f) ? v.x : a * v.x;
  v.y = v.y * di + b.y; v.y = (v.y >= 0.f) ? v.y : a * v.y;
  v.z = v.z * di + b.z; v.z = (v.z >= 0.f) ? v.z : a * v.z;
  v.w = v.w * di + b.w; v.w = (v.w >= 0.f) ? v.w : a * v.w;
  *reinterpret_cast<float4*>(Z + (size_t)row * 256 + c4) = v;
}

// Per-feature sum / sumsq over the batch dim; coalesced column reads.
__global__ __launch_bounds__(256) void bn_stats(const float* __restrict__ X,
                                                float* __restrict__ sum,
                                                float* __restrict__ sq, int n, int rpb) {
  const int f = threadIdx.x;
  const int r0 = blockIdx.x * rpb;
  const int r1 = min(r0 + rpb, n);
  float s = 0.f, q = 0.f;
  for (int r = r0; r < r1; ++r) {
    const float v = X[(size_t)r * 256 + f];
    s += v;
    q += v * v;
  }
  atomicAdd(&sum[f], s);
  atomicAdd(&sq[f], q);
}

__global__ __launch_bounds__(256) void bn_finalize(const float* __restrict__ sum,
                                                   const float* __restrict__ sq,
                                                   const float* __restrict__ gamma,
                                                   const float* __restrict__ beta,
                                                   float* __restrict__ scale,
                                                   float* __restrict__ shift, int n) {
  const int f = threadIdx.x;
  const float mean = sum[f] / (float)n;
  const float var = sq[f] / (float)n - mean * mean;  // biased (jnp.var default)
  const float s = gamma[f] * rsqrtf(var + 1e-5f);
  scale[f] = s;
  shift[f] = beta[f] - mean * s;
}

__global__ __launch_bounds__(256) void bn_apply(const float* __restrict__ X,
                                                const float* __restrict__ scale,
                                                const float* __restrict__ shift,
                                                const float* __restrict__ alpha_p,
                                                float* __restrict__ Y, int n,
                                                int use_prelu) {
  const int i = blockIdx.x * 256 + threadIdx.x;
  if (i >= n * 64) return;
  const int row = i >> 6;
  const int c4 = (i & 63) << 2;
  float4 v = *reinterpret_cast<const float4*>(X + (size_t)row * 256 + c4);
  const float4 sc = *reinterpret_cast<const float4*>(scale + c4);
  const float4 sh = *reinterpret_cast<const float4*>(shift + c4);
  v.x = v.x * sc.x + sh.x;
  v.y = v.y * sc.y + sh.y;
  v.z = v.z * sc.z + sh.z;
  v.w = v.w * sc.w + sh.w;
  if (use_prelu) {
    const float a = *alpha_p;
    v.x = (v.x >= 0.f) ? v.x : a * v.x;
    v.y = (v.y >= 0.f) ? v.y : a * v.y;
    v.z = (v.z >= 0.f) ? v.z : a * v.z;
    v.w = (v.w >= 0.f) ? v.w : a * v.w;
  }
  *reinterpret_cast<float4*>(Y + (size_t)row * 256 + c4) = v;
}

extern "C" void kernel_launch(void* const* d_in, const int* in_sizes, int n_in,
                              void* d_out, int out_size, void* d_ws, size_t ws_size,
                              hipStream_t stream) {
  (void)n_in; (void)out_size; (void)ws_size;
  const float* x        = (const float*)d_in[0];
  const int*   ei       = (const int*)d_in[1];
  const float* W1       = (const float*)d_in[2];
  const float* b1       = (const float*)d_in[3];
  const float* W2       = (const float*)d_in[4];
  const float* b2       = (const float*)d_in[5];
  const float* prelu_a  = (const float*)d_in[6];
  const float* bn_g     = (const float*)d_in[7];
  const float* bn_b     = (const float*)d_in[8];
  const float* projW    = (const float*)d_in[9];
  const float* projb    = (const float*)d_in[10];
  const float* pbn_g    = (const float*)d_in[11];
  const float* pbn_b    = (const float*)d_in[12];
  const float* prelu2_a = (const float*)d_in[13];

  const int n = in_sizes[0] / 256;   // N = 50000
  const int E = in_sizes[1] / 2;     // E = 1600000

  float* out   = (float*)d_out;
  float* z_out = out;                        // first half of d_out
  float* p_out = out + (size_t)n * 256;      // second half; doubles as z1/z2 temp
  float* T     = p_out;

  float* ws      = (float*)d_ws;
  float* G       = ws;                        // n*256
  float* ACC     = G + (size_t)n * 256;       // n*256
  float* deg     = ACC + (size_t)n * 256;     // n
  float* dinv    = deg + n;                   // n
  float* s_sum   = dinv + n;                  // 256
  float* s_sq    = s_sum + 256;               // 256 (contiguous with s_sum)
  float* s_scale = s_sq + 256;                // 256
  float* s_shift = s_scale + 256;             // 256
  float* PW1     = s_shift + 256;             // 65536 packed weights
  float* PW2     = PW1 + 65536;               // 65536
  float* PW3     = PW2 + 65536;               // 65536

  const int nb_n   = (n + 255) / 256;
  const int nb_e   = (E + 255) / 256;
  const int nb_el  = (n * 64 + 255) / 256;    // elementwise float4 grids
  const dim3 gemm_grid((n + 31) / 32);
  const int rpb = 256;
  const int nb_bn = (n + rpb - 1) / rpb;

  // One-time weight packing for b64 B-fragment loads.
  pack_w<<<128, 256, 0, stream>>>(W1, PW1);
  pack_w<<<128, 256, 0, stream>>>(W2, PW2);
  pack_w<<<128, 256, 0, stream>>>(projW, PW3);

  // Degrees (shared by both layers; self-loop contributes the initial 1.0)
  fill_f32<<<nb_n, 256, 0, stream>>>(deg, 1.0f, n);
  deg_kernel<<<nb_e, 256, 0, stream>>>(ei, deg, E);
  rsqrt_kernel<<<nb_n, 256, 0, stream>>>(deg, dinv, n);

  // Layer 1: g1 = (x@W1)*dinv[row]; ACC seeded with g1; scatter; epilogue
  gemm256<0><<<gemm_grid, 256, 0, stream>>>(x, PW1, dinv, G, ACC, n);
  scatter_kernel<<<(E + 3) / 4, 256, 0, stream>>>(G, ACC, ei, E);
  gcn_epilogue<<<nb_el, 256, 0, stream>>>(ACC, dinv, b1, prelu_a, T, n);

  // Layer 2
  gemm256<0><<<gemm_grid, 256, 0, stream>>>(T, PW2, dinv, G, ACC, n);
  scatter_kernel<<<(E + 3) / 4, 256, 0, stream>>>(G, ACC, ei, E);
  gcn_epilogue<<<nb_el, 256, 0, stream>>>(ACC, dinv, b2, prelu_a, T, n);

  // BatchNorm over z2 -> z (output 0)
  fill_f32<<<2, 256, 0, stream>>>(s_sum, 0.f, 512);
  bn_stats<<<nb_bn, 256, 0, stream>>>(T, s_sum, s_sq, n, rpb);
  bn_finalize<<<1, 256, 0, stream>>>(s_sum, s_sq, bn_g, bn_b, s_scale, s_shift, n);
  bn_apply<<<nb_el, 256, 0, stream>>>(T, s_scale, s_shift, prelu_a, z_out, n, 0);

  // Projection head: q = z@projW + projb; BN; PReLU -> p (output 1)
  gemm256<1><<<gemm_grid, 256, 0, stream>>>(z_out, PW3, projb, G, G, n);
  fill_f32<<<2, 256, 0, stream>>>(s_sum, 0.f, 512);
  bn_stats<<<nb_bn, 256, 0, stream>>>(G, s_sum, s_sq, n, rpb);
  bn_finalize<<<1, 256, 0, stream>>>(s_sum, s_sq, pbn_g, pbn_b, s_scale, s_shift, n);
  bn_apply<<<nb_el, 256, 0, stream>>>(G, s_scale, s_shift, prelu2_a, p_out, n, 1);
}